// RetinaNetRegressionLoss_12893491822714
// MI455X (gfx1250) — compile-verified
//
#include <hip/hip_runtime.h>

// RetinaNet regression loss, MI455X (gfx1250, wave32).
// Memory-bound streaming reduction: ~84 MB HBM traffic -> ~3.6 us floor at 23.3 TB/s.
// f32 math throughout (loss); WMMA f32 16x16x4 with all-ones B used as the
// exact cross-lane (wave32) sum engine.

#define A_COUNT 262144
#define B_COUNT 16
#define G_COUNT 128
#define TPB     256
#define ITEMS   8
#define BLK_X   (A_COUNT / (TPB * ITEMS))   // 128 blocks per image
#define WAVES   (TPB / 32)

typedef float v2f __attribute__((ext_vector_type(2)));
typedef float v8f __attribute__((ext_vector_type(8)));

// Exact 32-lane f32 sum via the matrix pipe.
// A layout (ISA 7.12.2, 32-bit A 16x4): lane L<16: VGPR0=A[L][0], VGPR1=A[L][1];
// lane L>=16: VGPR0=A[L-16][2], VGPR1=A[L-16][3]. With a.y=0 and B == all-ones
// (layout-independent), D[m][n] = v[m] + v[m+16] for every n. Each lane then
// holds 8 consecutive rows of D in its 8 VGPRs; summing them gives half-wave
// row sums, and one xor-16 shuffle completes the full reduction in all lanes.
__device__ __forceinline__ float wave_sum_wmma(float v) {
    v2f a; a.x = v;    a.y = 0.0f;
    v2f b; b.x = 1.0f; b.y = 1.0f;
    v8f c = {};
    v8f d = __builtin_amdgcn_wmma_f32_16x16x4_f32(
        /*neg_a=*/false, a, /*neg_b=*/false, b,
        /*c_mod=*/(short)0, c, /*reuse_a=*/false, /*reuse_b=*/false);
    float s = ((d[0] + d[1]) + (d[2] + d[3])) + ((d[4] + d[5]) + (d[6] + d[7]));
    s += __shfl_xor(s, 16, 32);
    return s;
}

__global__ __launch_bounds__(TPB)
void retina_loss_partials(const float4* __restrict__ reg,      // [B, A] float4
                          const float4* __restrict__ anchors,  // [A] float4
                          const float4* __restrict__ gt,       // [B, G] float4
                          const int*    __restrict__ midx,     // [B, A]
                          float* __restrict__ loss_part,       // [B, BLK_X]
                          float* __restrict__ cnt_part) {      // [B, BLK_X]
    __shared__ float4 sGt[G_COUNT];      // per-image gt table: 2 KB, gather via LDS
    __shared__ float  sWL[WAVES], sWC[WAVES];

    const int b = blockIdx.y;
    const int t = threadIdx.x;

    if (t < G_COUNT) sGt[t] = gt[b * G_COUNT + t];
    __syncthreads();

    const size_t imgBase = (size_t)b * A_COUNT;
    const int    base    = blockIdx.x * (TPB * ITEMS);

    float lsum = 0.0f, csum = 0.0f;
#pragma unroll
    for (int i = 0; i < ITEMS; ++i) {
        const int ai = base + i * TPB + t;           // coalesced b128 across lanes
        const int m  = midx[imgBase + ai];
        const float4 an = anchors[ai];
        const float4 rg = reg[imgBase + ai];
        const float4 g  = sGt[m < 0 ? 0 : m];        // clamped gather (matches ref)

        const float ex_w  = an.z - an.x;
        const float ex_h  = an.w - an.y;
        const float ex_cx = an.x + 0.5f * ex_w;
        const float ex_cy = an.y + 0.5f * ex_h;
        const float gw    = g.z - g.x;
        const float gh    = g.w - g.y;
        const float gcx   = g.x + 0.5f * gw;
        const float gcy   = g.y + 0.5f * gh;

        const float dx = (gcx - ex_cx) / ex_w;
        const float dy = (gcy - ex_cy) / ex_h;
        const float dw = logf(gw / ex_w);
        const float dh = logf(gh / ex_h);

        const float l1 = fabsf(rg.x - dx) + fabsf(rg.y - dy) +
                         fabsf(rg.z - dw) + fabsf(rg.w - dh);
        if (m >= 0) { lsum += l1; csum += 1.0f; }
    }

    // Wave reductions: unconditional, full block -> EXEC all-ones (WMMA req.)
    lsum = wave_sum_wmma(lsum);
    csum = wave_sum_wmma(csum);

    if ((t & 31) == 0) { sWL[t >> 5] = lsum; sWC[t >> 5] = csum; }
    __syncthreads();
    if (t == 0) {
        float L = 0.0f, C = 0.0f;
#pragma unroll
        for (int w = 0; w < WAVES; ++w) { L += sWL[w]; C += sWC[w]; }
        loss_part[b * BLK_X + blockIdx.x] = L;
        cnt_part [b * BLK_X + blockIdx.x] = C;
    }
}

__global__ __launch_bounds__(TPB)
void retina_loss_finalize(const float* __restrict__ loss_part,
                          const float* __restrict__ cnt_part,
                          float* __restrict__ out) {
    __shared__ float sWL[WAVES], sWC[WAVES];
    const int t = threadIdx.x;
    float total = 0.0f;

    for (int b = 0; b < B_COUNT; ++b) {
        float l = (t < BLK_X) ? loss_part[b * BLK_X + t] : 0.0f;
        float c = (t < BLK_X) ? cnt_part [b * BLK_X + t] : 0.0f;
        l = wave_sum_wmma(l);
        c = wave_sum_wmma(c);
        if ((t & 31) == 0) { sWL[t >> 5] = l; sWC[t >> 5] = c; }
        __syncthreads();
        if (t == 0) {
            float L = 0.0f, C = 0.0f;
#pragma unroll
            for (int w = 0; w < WAVES; ++w) { L += sWL[w]; C += sWC[w]; }
            total += L / fmaxf(1.0f, C);
        }
        __syncthreads();
    }
    if (t == 0) out[0] = total / (float)B_COUNT;
}

extern "C" void kernel_launch(void* const* d_in, const int* in_sizes, int n_in,
                              void* d_out, int out_size, void* d_ws, size_t ws_size,
                              hipStream_t stream) {
    const float4* reg     = (const float4*)d_in[0];   // bbox_regression [B,A,4] f32
    const float4* anchors = (const float4*)d_in[1];   // anchors [A,4] f32
    const float4* gt      = (const float4*)d_in[2];   // gt_boxes [B,G,4] f32
    const int*    midx    = (const int*)d_in[3];      // matched_idxs [B,A] i32
    float*        out     = (float*)d_out;            // scalar f32

    float* loss_part = (float*)d_ws;                  // [B, BLK_X]
    float* cnt_part  = loss_part + B_COUNT * BLK_X;   // [B, BLK_X]  (16 KB total)

    dim3 grid(BLK_X, B_COUNT);
    retina_loss_partials<<<grid, TPB, 0, stream>>>(reg, anchors, gt, midx,
                                                   loss_part, cnt_part);
    retina_loss_finalize<<<1, TPB, 0, stream>>>(loss_part, cnt_part, out);
}